// DeepWalk_44727789421079
// MI455X (gfx1250) — compile-verified
//
#include <hip/hip_runtime.h>
#include <stdint.h>

// ---------------- CDNA5 TDM descriptor vector types (probe-verified arity) ---
typedef unsigned int u32x4 __attribute__((ext_vector_type(4)));
typedef int          i32x8 __attribute__((ext_vector_type(8)));
typedef int          i32x4 __attribute__((ext_vector_type(4)));

#define WALK_LEN   80
#define TILE_ROWS  128          // nodes per block == blockDim.x
#define ROW_PAD_DW 85           // 80 data dwords + 5 pad dwords (1 per 16)
#define LDS_DW     (TILE_ROWS * ROW_PAD_DW)   // 10880 dwords = 43520 B

__global__ __launch_bounds__(TILE_ROWS)
void deepwalk_kernel(const int* __restrict__ indptr,
                     const int* __restrict__ indices,
                     const float* __restrict__ rand_vals,
                     float* __restrict__ out,
                     int N, int W)
{
    __shared__ float lds_rand[LDS_DW];

    const int t  = threadIdx.x;
    const int n0 = blockIdx.x * TILE_ROWS;
    const int w  = blockIdx.y;

    // ---- Wave 0 issues the Tensor Data Mover load: 128x80 f32 tile of
    // rand_vals[w, n0:n0+128, 0:80] (row stride = 80 f32 = 320 B) packed into
    // LDS with 1-dword padding every 16 dwords (bank-conflict-free reads).
    if (t < 32) {   // wave-uniform branch; TDM ignores EXEC so must branch, not predicate
        const uint64_t gaddr =
            (uint64_t)(uintptr_t)(rand_vals + ((uint64_t)w * (uint64_t)N + (uint64_t)n0) * WALK_LEN);
        const uint32_t lds_base = (uint32_t)(uintptr_t)&lds_rand[0]; // low 32b of flat = LDS offset
        const uint32_t rows_rem = (uint32_t)(N - n0);                // OOB rows read as zero

        // ---- D# group 0 (128b): count=1 | lds_addr | global_addr | type=2
        u32x4 g0;
        g0[0] = 1u;                                   // count=1, is_restore=0, gather=0
        g0[1] = lds_base;                             // lds_addr [63:32]
        g0[2] = (uint32_t)(gaddr & 0xFFFFFFFFu);      // global_addr [95:64]
        g0[3] = (uint32_t)((gaddr >> 32) & 0x01FFFFFFu) | 0x80000000u; // addr[56:32] | type=2

        // ---- D# group 1 (256b)
        // d0: wg_mask=0 | data_size=2(4B) | pad_enable | pad_interval=3(16dw) | pad_amount=0(1dw)
        i32x8 g1;
        g1[0] = (int)((2u << 16) | (1u << 20) | (3u << 22));          // 0x00D20000
        g1[1] = (int)(((uint32_t)WALK_LEN & 0xFFFFu) << 16);          // tensor_dim0[15:0] @ [63:48]
        g1[2] = (int)(((uint32_t)WALK_LEN >> 16) |
                      ((rows_rem & 0xFFFFu) << 16));                  // tdim0 hi | tensor_dim1 lo
        g1[3] = (int)((rows_rem >> 16) |
                      ((uint32_t)WALK_LEN << 16));                    // tensor_dim1 hi | tile_dim0=80
        g1[4] = (int)((uint32_t)TILE_ROWS);                           // tile_dim1=128 | tile_dim2=0
        g1[5] = (int)((uint32_t)WALK_LEN);                            // tensor_dim0_stride = 80 elems
        g1[6] = 0;                                                    // stride0 hi | stride1 lo (2D)
        g1[7] = 0;

        i32x4 g2 = {0, 0, 0, 0};     // 2D tensor: groups 2/3 unused
        i32x4 g3 = {0, 0, 0, 0};

#if defined(__clang_major__) && (__clang_major__ >= 23)
        i32x8 gx = {0, 0, 0, 0, 0, 0, 0, 0};
        __builtin_amdgcn_tensor_load_to_lds(g0, g1, g2, g3, gx, 0);
#else
        __builtin_amdgcn_tensor_load_to_lds(g0, g1, g2, g3, 0);
#endif
        __builtin_amdgcn_s_wait_tensorcnt((short)0);  // TENSORcnt==0: tile is in LDS
    }
    __syncthreads();

    const int n = n0 + t;
    if (n >= N) return;

    int  curr  = n;
    bool alive = true;
    const int lbase = t * ROW_PAD_DW;
    float4* __restrict__ out4 =
        reinterpret_cast<float4*>(out + ((size_t)w * (size_t)N + (size_t)n) * WALK_LEN);

    // one serial-gather step; u is passed in (loaded off the critical path)
    auto step = [&](float u) -> float {
        const float rec = alive ? (float)curr : -1.0f;
        const int  rs  = indptr[curr];
        const int  deg = indptr[curr + 1] - rs;        // two parallel L2 loads
        const bool has = deg > 0;
        int off = (int)(u * (float)deg);
        const int mx = (deg - 1) > 0 ? (deg - 1) : 0;
        off = off < 0 ? 0 : (off > mx ? mx : off);
        const int nxt = indices[rs + off];             // dependent L2 gather
        curr  = (alive && has) ? nxt : curr;
        alive = alive && has;
        return rec;
    };

    for (int lc = 0; lc < WALK_LEN; lc += 4) {
        // hoist the 4 LDS reads: independent of the chain, pipeline as ds_load_b32 x4
        const float u0 = lds_rand[lbase + (lc + 0) + ((lc + 0) >> 4)];
        const float u1 = lds_rand[lbase + (lc + 1) + ((lc + 1) >> 4)];
        const float u2 = lds_rand[lbase + (lc + 2) + ((lc + 2) >> 4)];
        const float u3 = lds_rand[lbase + (lc + 3) + ((lc + 3) >> 4)];

        float4 r4;
        r4.x = step(u0);
        r4.y = step(u1);
        r4.z = step(u2);
        r4.w = step(u3);

        out4[lc >> 2] = r4;   // global_store_b128; lines merge in L2 before DRAM
    }
}

extern "C" void kernel_launch(void* const* d_in, const int* in_sizes, int n_in,
                              void* d_out, int out_size, void* d_ws, size_t ws_size,
                              hipStream_t stream) {
    const int*   indptr    = (const int*)d_in[0];
    const int*   indices   = (const int*)d_in[1];
    const float* rand_vals = (const float*)d_in[2];
    float*       out       = (float*)d_out;

    const int N = in_sizes[0] - 1;                      // indptr has N+1 entries
    const int W = in_sizes[2] / (N * WALK_LEN);         // rand_vals is [W, N, L]

    dim3 grid((N + TILE_ROWS - 1) / TILE_ROWS, W);
    dim3 block(TILE_ROWS);
    deepwalk_kernel<<<grid, block, 0, stream>>>(indptr, indices, rand_vals, out, N, W);
}